// ES_Normalize_44968307589367
// MI455X (gfx1250) — compile-verified
//
#include <hip/hip_runtime.h>

// EMA scan y[t] = (1-a)*x[t] + a*y[t-1], y[0]=x[0], over (B=8, T=4096, D=1024) fp32.
// Memory-bound (~272 MiB incl. warm-up => ~11.7us @ 23.3 TB/s).
// Parallelism: alpha^33 ~ 5e-18 < fp32 ulp, so each 256-step time chunk restarts
// from zero carry 32 steps early (warm-up) => all 8192 (b, d-tile, chunk) waves
// independent. Within a 16-step block the scan is a lower-triangular matmul done
// with 4 chained V_WMMA_F32_16X16X4_F32; the carry row is broadcast across lanes
// with ds_bpermute_b32. Addressing: one per-lane base pointer per stream,
// advanced 64KB/block, all row offsets folded into 24-bit immediate offsets.

typedef __attribute__((ext_vector_type(2))) float v2f;
typedef __attribute__((ext_vector_type(8))) float v8f;

#define EMA_ALPHA 0.3f

constexpr int kB = 8;
constexpr int kT = 4096;
constexpr int kD = 1024;
constexpr int kChunk   = 256;            // timesteps per wave (16 blocks of 16)
constexpr int kBlocks  = kChunk / 16;    // 16 written blocks per wave
constexpr int kNChunk  = kT / kChunk;    // 16
constexpr int kWarmBlk = 2;              // 32 warm-up steps: alpha^33 ~ 5e-18
constexpr int kDTiles  = kD / 16;        // 64 channel tiles
constexpr int kWaves   = kB * kNChunk * kDTiles;  // 8192 waves
constexpr int kWavesPerBlock = 8;        // 256 threads / wave32

__global__ __launch_bounds__(256) void ema_scan_wmma_kernel(
    const float* __restrict__ x, float* __restrict__ out) {
  const int lane = threadIdx.x & 31;
  const int ln   = lane & 15;
  const bool hi  = lane >= 16;
  const int wid  = blockIdx.x * (blockDim.x >> 5) + (threadIdx.x >> 5);

  const int dtile = wid % kDTiles;
  const int chunk = (wid / kDTiles) % kNChunk;
  const int b     = wid / (kDTiles * kNChunk);

  const int col = dtile * 16 + ln;  // channel this lane owns (N index)

  // ---- A operand: M[m][k] = (k<=m) ? (1-a)*a^(m-k) : 0, m = ln.
  // 16x4 f32 A layout: V0 = K = 4kk + (hi?2:0), V1 = next K.
  float Avals[4][2];
#pragma unroll
  for (int kk = 0; kk < 4; ++kk) {
#pragma unroll
    for (int j = 0; j < 2; ++j) {
      const int k = 4 * kk + (hi ? 2 : 0) + j;
      float v = 0.0f;
      if (k <= ln) {
        v = 1.0f - EMA_ALPHA;
        for (int e = 0; e < ln - k; ++e) v *= EMA_ALPHA;
      }
      Avals[kk][j] = v;
    }
  }

  // ---- carry-injection powers: C_init[row i][n] = a^(i+1)*c[n];
  // C/D layout: VGPR r holds row i = r + (hi?8:0).
  float vpow[8];
  {
    float p = EMA_ALPHA;
    if (hi) {
#pragma unroll
      for (int e = 0; e < 8; ++e) p *= EMA_ALPHA;
    }
#pragma unroll
    for (int r = 0; r < 8; ++r) { vpow[r] = p; p *= EMA_ALPHA; }
  }

  const int warm    = (chunk == 0) ? 0 : kWarmBlk;
  const int t_start = chunk * kChunk - warm * 16;

  // Per-lane stream pointers; all block-internal rows are immediate offsets.
  const float* px = x   + ((size_t)b * kT + t_start + (hi ? 2 : 0)) * kD + col;
  float*       po = out + ((size_t)b * kT + chunk * kChunk + (hi ? 8 : 0)) * kD + col;

  // Initial carry: chunk 0 uses c = x[b,0,n] (gives y[0]=x[0] to 1 ulp, fixed
  // exactly below); other chunks warm up from zero.
  const float c0 = (chunk == 0) ? x[(size_t)b * kT * kD + col] : 0.0f;
  float c = c0;

  // B operand: 4x16 f32, layout symmetric to A: V0 = row 4kk+(hi?2:0), V1 = +1.
  auto load_block = [&](const float* p, v2f bb[4]) {
#pragma unroll
    for (int kk = 0; kk < 4; ++kk) {
      bb[kk].x = p[(size_t)(4 * kk) * kD];
      bb[kk].y = p[(size_t)(4 * kk + 1) * kD];
    }
  };

  // One 16x16 block: acc = M*X + v(x)c via 4 chained fp32 WMMAs; update carry.
  auto compute = [&](const v2f bb[4]) -> v8f {
    v8f acc;
#pragma unroll
    for (int r = 0; r < 8; ++r) acc[r] = vpow[r] * c;
#pragma unroll
    for (int kk = 0; kk < 4; ++kk) {
      v2f aop;
      aop.x = Avals[kk][0];
      aop.y = Avals[kk][1];
      acc = __builtin_amdgcn_wmma_f32_16x16x4_f32(
          false, aop, false, bb[kk], (short)0, acc, false, false);
    }
    // Carry broadcast: row 15 = acc[7] of lanes 16..31 -> all lanes, index ln.
    c = __int_as_float(
        __builtin_amdgcn_ds_bpermute((16 + ln) * 4, __float_as_int(acc[7])));
    return acc;
  };

  v2f cur[4], nxt[4];
  load_block(px, cur);

  // Warm-up blocks (chunk > 0 only): compute, no stores.
  for (int w = 0; w < warm; ++w) {
    px += 16 * kD;
    load_block(px, nxt);
    (void)compute(cur);
#pragma unroll
    for (int kk = 0; kk < 4; ++kk) cur[kk] = nxt[kk];
  }

  // Main blocks: double-buffered loads, non-temporal streaming stores.
  for (int blk = 0; blk < kBlocks; ++blk) {
    const bool more = (blk + 1 < kBlocks);
    if (more) {
      px += 16 * kD;
      load_block(px, nxt);
    }
    v8f acc = compute(cur);
#pragma unroll
    for (int r = 0; r < 8; ++r) {
      __builtin_nontemporal_store(acc[r], po + (size_t)r * kD);
    }
    po += 16 * kD;
    if (more) {
#pragma unroll
      for (int kk = 0; kk < 4; ++kk) cur[kk] = nxt[kk];
    }
  }

  // Restore y[0] = x[0] bit-exactly (same-lane same-address stores are ordered).
  if (chunk == 0 && lane < 16) {
    out[(size_t)b * kT * kD + col] = c0;
  }
  // Second output: out[:, -1, :] appended after the main tensor; c == y[T-1][n].
  if (chunk == kNChunk - 1 && lane < 16) {
    out[(size_t)kB * kT * kD + (size_t)b * kD + col] = c;
  }
}

extern "C" void kernel_launch(void* const* d_in, const int* in_sizes, int n_in,
                              void* d_out, int out_size, void* d_ws, size_t ws_size,
                              hipStream_t stream) {
  (void)in_sizes; (void)n_in; (void)out_size; (void)d_ws; (void)ws_size;
  const float* x = (const float*)d_in[0];
  float* out = (float*)d_out;
  dim3 grid(kWaves / kWavesPerBlock);   // 1024 blocks
  dim3 block(kWavesPerBlock * 32);      // 256 threads = 8 wave32
  ema_scan_wmma_kernel<<<grid, block, 0, stream>>>(x, out);
}